// WaveGRU_2619930050860
// MI455X (gfx1250) — compile-verified
//
#include <hip/hip_runtime.h>

// ---------------------------------------------------------------------------
// WaveGRU for MI455X (gfx1250): fp32 WMMA (V_WMMA_F32_16X16X4_F32) throughout.
//   Phase 1: gx = layer_in @ Wi^T + bi        (big parallel GEMM, WMMA)
//   Phase 2: persistent grid-resident recurrence, Wh cached in LDS,
//            dual WMMA accumulate chains, gx prefetch, 1 barrier/step.
// ---------------------------------------------------------------------------

typedef __attribute__((ext_vector_type(2))) float v2f;
typedef __attribute__((ext_vector_type(8))) float v8f;

#define DEV __device__ __forceinline__

constexpr int Hdim = 1024;            // hidden size
constexpr int Ddim = 1024;            // input size
constexpr int Bsz  = 32;              // batch
constexpr int Slen = 1024;            // sequence length
constexpr int GXC  = 3 * Hdim;        // 3072 gate columns
constexpr int NBLK = Hdim / 16;       // 64 persistent blocks in phase 2

// LDS layout for phase 2 (dynamic shared memory):
//   Wh slice: 48 rows (3 gates x 16 cols) x 1028 floats (1024 + 4 pad: the
//   pad makes lane n hit bank 4n, so 16-lane x b64 B-fragment loads cover all
//   64 banks conflict-free)
constexpr int WROW = 1028;
constexpr int LDSW_FLOATS = 48 * WROW;                 // 49344
constexpr int LDSP_FLOATS = 6 * 4 * 16 * 16;           // 6144 partials
constexpr size_t SMEM_BYTES = (size_t)(LDSW_FLOATS + LDSP_FLOATS) * 4; // ~217KB

DEV v8f wmma4(v2f a, v2f b, v8f c) {
    // D(16x16,f32) += A(16x4,f32) * B(4x16,f32) ; 8-arg CDNA5 form
    return __builtin_amdgcn_wmma_f32_16x16x4_f32(
        /*neg_a=*/false, a, /*neg_b=*/false, b,
        /*c_mod=*/(short)0, c, /*reuse_a=*/false, /*reuse_b=*/false);
}

DEV float sigmoidf_(float x) { return 1.0f / (1.0f + __expf(-x)); }

// ---------------------------------------------------------------------------
// Phase 1: gx[(b*S+s), g*H+j] = sum_k A[(b*S+s),k] * W[(g*H+j),k] + bias[g*H+j]
// grid = (M/16, GXC/512), block = 256 (8 waves); each wave -> 16x64 strip.
// ---------------------------------------------------------------------------
__global__ __launch_bounds__(256) void gru_input_gemm(
    const float* __restrict__ A,     // (B*S, D)
    const float* __restrict__ W,     // (3H, D) row-major
    const float* __restrict__ bias,  // (3H)
    float* __restrict__ gx)          // (B*S, 3H)
{
    const int lane = threadIdx.x & 31;
    const int wave = threadIdx.x >> 5;                 // 0..7
    const int m0   = blockIdx.x * 16;
    const int n0   = blockIdx.y * 512 + wave * 64;
    const int m    = lane & 15;
    const int n    = lane & 15;
    const int kb   = (lane >> 4) << 1;                 // 0 or 2 (K sub-pair)

    const float* Arow = A + (size_t)(m0 + m) * Ddim + kb;
    const float* W0   = W + (size_t)(n0 +  0 + n) * Ddim + kb;
    const float* W1   = W + (size_t)(n0 + 16 + n) * Ddim + kb;
    const float* W2   = W + (size_t)(n0 + 32 + n) * Ddim + kb;
    const float* W3   = W + (size_t)(n0 + 48 + n) * Ddim + kb;

    float b0 = bias[n0 +  0 + n];
    float b1 = bias[n0 + 16 + n];
    float b2 = bias[n0 + 32 + n];
    float b3 = bias[n0 + 48 + n];
    v8f c0 = {b0,b0,b0,b0,b0,b0,b0,b0};
    v8f c1 = {b1,b1,b1,b1,b1,b1,b1,b1};
    v8f c2 = {b2,b2,b2,b2,b2,b2,b2,b2};
    v8f c3 = {b3,b3,b3,b3,b3,b3,b3,b3};

    #pragma unroll 4
    for (int k = 0; k < Ddim; k += 4) {
        v2f a = *(const v2f*)(Arow + k);
        c0 = wmma4(a, *(const v2f*)(W0 + k), c0);
        c1 = wmma4(a, *(const v2f*)(W1 + k), c1);
        c2 = wmma4(a, *(const v2f*)(W2 + k), c2);
        c3 = wmma4(a, *(const v2f*)(W3 + k), c3);
    }

    const int rbase = m0 + ((lane >> 4) << 3);         // rows r..r+7 or r+8..r+15
    #pragma unroll
    for (int r = 0; r < 8; ++r) {
        size_t row = (size_t)(rbase + r);
        gx[row * GXC + n0 +  0 + n] = c0[r];
        gx[row * GXC + n0 + 16 + n] = c1[r];
        gx[row * GXC + n0 + 32 + n] = c2[r];
        gx[row * GXC + n0 + 48 + n] = c3[r];
    }
}

// ---------------------------------------------------------------------------
// Software grid barrier (sense via monotonically increasing generation).
// ---------------------------------------------------------------------------
DEV void grid_barrier(unsigned* cnt, unsigned* gen, unsigned nb) {
    __syncthreads();
    if (threadIdx.x == 0) {
        __threadfence();
        unsigned g = atomicAdd(gen, 0u);               // acquire-ish read
        if (atomicAdd(cnt, 1u) == nb - 1u) {
            atomicExch(cnt, 0u);
            __threadfence();
            atomicAdd(gen, 1u);                        // release this round
        } else {
            while (atomicAdd(gen, 0u) == g) { __builtin_amdgcn_s_sleep(1); }
        }
        __threadfence();
    }
    __syncthreads();
}

// ---------------------------------------------------------------------------
// Phase 2: persistent recurrence. 64 blocks (one per 16-wide H tile),
// 768 threads = 24 waves: tile = gate*2+mtile (6 tiles) x 4 K-slices.
// Wh slice lives in LDS for all 1024 steps; h ping-pongs through L2.
// ---------------------------------------------------------------------------
__global__ __launch_bounds__(768) void gru_recurrent(
    const float* __restrict__ gx,    // (B*S, 3H), bi already folded in
    const float* __restrict__ Wh,    // (3, H, H) row-major
    const float* __restrict__ bh,    // (3H)
    const float* __restrict__ h0,    // (B, H)
    float* __restrict__ out,         // (B, S, H)
    float* __restrict__ hfinal,      // (B, H)
    float* __restrict__ hbuf0,       // (B, H) double buffer
    float* __restrict__ hbuf1,
    unsigned* __restrict__ bar_cnt,
    unsigned* __restrict__ bar_gen)
{
    extern __shared__ float smem[];
    float* ldsW = smem;                                // 48 x 1028 Wh slice
    float* ldsP = smem + LDSW_FLOATS;                  // 6x4x16x16 partials

    const int tid   = threadIdx.x;
    const int lane  = tid & 31;
    const int wave  = tid >> 5;                        // 0..23
    const int tile  = wave >> 2;                       // 0..5 = gate*2+mtile
    const int ks    = wave & 3;                        // K slice 0..3
    const int gate  = tile >> 1;
    const int mtile = tile & 1;
    const int j0    = blockIdx.x * 16;                 // this block's H columns

    const int m  = lane & 15;
    const int n  = lane & 15;
    const int kb = (lane >> 4) << 1;
    const int b0 = mtile * 16;
    const int k0 = ks * 256;

    // --- preload this WG's Wh slice into LDS (rows: gate*16 + col) ---------
    for (int idx = tid; idx < 3 * 16 * Hdim; idx += 768) {
        int k   = idx & (Hdim - 1);
        int row = idx >> 10;                           // 0..47
        int g   = row >> 4, nn = row & 15;
        ldsW[row * WROW + k] =
            Wh[(size_t)g * Hdim * Hdim + (size_t)(j0 + nn) * Hdim + k];
    }
    // --- init h buffer 0 from h0 (each block owns its 16 columns) ----------
    if (tid < Bsz * 16) {
        int b = tid >> 4, nn = tid & 15;
        hbuf0[b * Hdim + j0 + nn] = h0[b * Hdim + j0 + nn];
    }
    grid_barrier(bar_cnt, bar_gen, NBLK);              // also covers LDS fill

    // B fragment source in LDS: B[k][n] = Wh[gate][(j0+n)][k]
    const float* Bp = ldsW + (gate * 16 + n) * WROW + k0 + kb;

    for (int t = 0; t < Slen; ++t) {
        const float* hc  = (t & 1) ? hbuf1 : hbuf0;    // read
        float*       hnx = (t & 1) ? hbuf0 : hbuf1;    // write

        // prefetch next step's gx rows (hides HBM latency in the serial chain)
        if (tid < Bsz * 16) {
            int b = tid >> 4, nn = tid & 15;
            int tn = (t + 1 < Slen) ? (t + 1) : t;
            const float* gp = gx + ((size_t)b * Slen + tn) * GXC + j0 + nn;
            __builtin_prefetch(gp, 0, 0);
            __builtin_prefetch(gp + Hdim, 0, 0);
            __builtin_prefetch(gp + 2 * Hdim, 0, 0);
        }

        const float* Ap = hc + (size_t)(b0 + m) * Hdim + k0 + kb;
        // two independent accumulate chains (depth 32 each instead of 64)
        v8f acc0 = {0.f,0.f,0.f,0.f,0.f,0.f,0.f,0.f};
        v8f acc1 = {0.f,0.f,0.f,0.f,0.f,0.f,0.f,0.f};
        #pragma unroll 4
        for (int k = 0; k < 128; k += 4) {
            v2f aA = *(const v2f*)(Ap + k);
            v2f aB = *(const v2f*)(Bp + k);            // LDS (ds_load_b64)
            acc0 = wmma4(aA, aB, acc0);
            v2f bA = *(const v2f*)(Ap + 128 + k);
            v2f bB = *(const v2f*)(Bp + 128 + k);
            acc1 = wmma4(bA, bB, acc1);
        }
        v8f acc = acc0 + acc1;

        // stash partial 16x16 tile
        {
            int mb = (lane >> 4) << 3;
            float* dst = &ldsP[(((tile * 4 + ks) * 16) + mb) * 16 + n];
            #pragma unroll
            for (int r = 0; r < 8; ++r) dst[r * 16] = acc[r];
        }
        __syncthreads();

        if (tid < Bsz * 16) {                          // 512 gate lanes
            int b  = tid >> 4, nn = tid & 15;
            int mt = b >> 4,  mm = b & 15;
            int j  = j0 + nn;
            float hr = bh[0 * Hdim + j];
            float hz = bh[1 * Hdim + j];
            float hn = bh[2 * Hdim + j];
            #pragma unroll
            for (int s = 0; s < 4; ++s) {
                hr += ldsP[(((0 * 2 + mt) * 4 + s) * 16 + mm) * 16 + nn];
                hz += ldsP[(((1 * 2 + mt) * 4 + s) * 16 + mm) * 16 + nn];
                hn += ldsP[(((2 * 2 + mt) * 4 + s) * 16 + mm) * 16 + nn];
            }
            size_t rb = (size_t)b * Slen + (size_t)t;
            const float* gxr = gx + rb * GXC;
            float r  = sigmoidf_(gxr[0 * Hdim + j] + hr);
            float z  = sigmoidf_(gxr[1 * Hdim + j] + hz);
            float nv = tanhf(gxr[2 * Hdim + j] + r * hn);
            float hp = hc[b * Hdim + j];
            float hnew = (1.0f - z) * nv + z * hp;
            hnx[b * Hdim + j] = hnew;
            out[rb * Hdim + j] = hnew;
            if (t == Slen - 1) hfinal[b * Hdim + j] = hnew;
        }
        grid_barrier(bar_cnt, bar_gen, NBLK);          // leading __syncthreads
    }                                                  // also protects LDS reuse
}

// ---------------------------------------------------------------------------
// Launch: layer 0 (gemm + recurrence) then layer 1.
// Inputs: x, h0, Wi, Wh, bi, bh (all fp32).
// Output: [final layer output (B,S,H)] ++ [h_final (L,B,H)].
// ---------------------------------------------------------------------------
extern "C" void kernel_launch(void* const* d_in, const int* in_sizes, int n_in,
                              void* d_out, int out_size, void* d_ws, size_t ws_size,
                              hipStream_t stream) {
    (void)in_sizes; (void)n_in; (void)out_size; (void)ws_size;
    const float* x  = (const float*)d_in[0];
    const float* h0 = (const float*)d_in[1];
    const float* Wi = (const float*)d_in[2];
    const float* Wh = (const float*)d_in[3];
    const float* bi = (const float*)d_in[4];
    const float* bh = (const float*)d_in[5];
    float* out = (float*)d_out;

    char* ws = (char*)d_ws;
    unsigned* bar = (unsigned*)ws;                              // 256 B reserved
    float* gxbuf  = (float*)(ws + 256);                         // B*S*3H fp32
    float* mid    = gxbuf + (size_t)Bsz * Slen * GXC;           // B*S*H fp32
    float* hb0    = mid   + (size_t)Bsz * Slen * Hdim;          // B*H
    float* hb1    = hb0   + (size_t)Bsz * Hdim;                 // B*H

    hipMemsetAsync(d_ws, 0, 256, stream);                       // barrier state

    float* hfin = out + (size_t)Bsz * Slen * Hdim;              // (L,B,H)

    dim3 ggrid(Bsz * Slen / 16, GXC / 512);                     // (2048, 6)

    // ---- layer 0 ----
    gru_input_gemm<<<ggrid, 256, 0, stream>>>(x, Wi, bi, gxbuf);
    gru_recurrent<<<NBLK, 768, SMEM_BYTES, stream>>>(gxbuf, Wh, bh, h0,
                                                     mid, hfin, hb0, hb1,
                                                     bar, bar + 1);
    // ---- layer 1 ----
    gru_input_gemm<<<ggrid, 256, 0, stream>>>(mid, Wi + (size_t)3 * Hdim * Ddim,
                                              bi + 3 * Hdim, gxbuf);
    gru_recurrent<<<NBLK, 768, SMEM_BYTES, stream>>>(gxbuf,
                                                     Wh + (size_t)3 * Hdim * Hdim,
                                                     bh + 3 * Hdim,
                                                     h0 + Bsz * Hdim,
                                                     out, hfin + Bsz * Hdim,
                                                     hb0, hb1, bar, bar + 1);
}